// MatchHistogram_15066745275095
// MI455X (gfx1250) — compile-verified
//
#include <hip/hip_runtime.h>
#include <stdint.h>

#define NBINS 256

typedef float v2f __attribute__((ext_vector_type(2)));
typedef float v8f __attribute__((ext_vector_type(8)));

// Quantize exactly like the reference: clip(floor((s+1)/2*255), 0, 255)
__device__ __forceinline__ int bin_of(float s) {
    float t = __builtin_floorf((s + 1.0f) * 0.5f * 255.0f);
    t = fminf(fmaxf(t, 0.0f), 255.0f);
    return (int)t;
}

// ---------------------------------------------------------------------------
// Kernel 0: zero the 256-bin global histogram (d_ws is poisoned by harness)
// ---------------------------------------------------------------------------
__global__ void __launch_bounds__(256) hist_zero_kernel(unsigned int* __restrict__ g) {
    g[threadIdx.x] = 0u;
}

// ---------------------------------------------------------------------------
// Kernel 1: streaming histogram. float4 loads, 8 per-wave LDS sub-histograms
// (ds_add_u32), block reduce, one global atomic per bin per block.
// ---------------------------------------------------------------------------
__global__ void __launch_bounds__(256) hist_kernel(const float* __restrict__ src,
                                                   unsigned int* __restrict__ ghist,
                                                   int n4) {
    __shared__ unsigned int sh[8][NBINS];
    const int t    = threadIdx.x;
    const int wave = t >> 5;                 // wave32: 8 waves per 256-thread block

    for (int i = t; i < 8 * NBINS; i += 256) (&sh[0][0])[i] = 0u;
    __syncthreads();

    unsigned int* myh = sh[wave];
    const float4* s4  = (const float4*)src;
    const int stride  = gridDim.x * 256;

    for (int i = blockIdx.x * 256 + t; i < n4; i += stride) {
        int ip = i + stride;
        if (ip < n4) __builtin_prefetch(&s4[ip], 0, 0);   // global_prefetch_b8
        float4 v = s4[i];
        atomicAdd(&myh[bin_of(v.x)], 1u);
        atomicAdd(&myh[bin_of(v.y)], 1u);
        atomicAdd(&myh[bin_of(v.z)], 1u);
        atomicAdd(&myh[bin_of(v.w)], 1u);
    }
    __syncthreads();

    unsigned int sum = 0;
    #pragma unroll
    for (int w = 0; w < 8; ++w) sum += sh[w][t];
    if (sum) atomicAdd(&ghist[t], sum);
}

// ---------------------------------------------------------------------------
// Kernel 2: one wave32 builds the LUT.
// 256-element inclusive scan as S = L(16x16 lower-tri ones) x H(16x16),
// chained over four V_WMMA_F32_16X16X4_F32 (K=16 = 4 slices of K=4).
//   H[r][seg] = hist[seg*16 + r]  ->  S[r][seg] = within-segment scan.
// Then segment offsets + normalization, normal CDF via erff, searchsorted-left
// by counting, and the final /255*2-1 rescale folded into the stored LUT.
// ---------------------------------------------------------------------------
__global__ void __launch_bounds__(32) scan_lut_kernel(const unsigned int* __restrict__ ghist,
                                                      float* __restrict__ lut) {
    __shared__ float hf[NBINS];
    __shared__ float S[NBINS];
    __shared__ float nc[NBINS];

    const int lane = threadIdx.x;            // single wave32, EXEC all-1s for WMMA
    for (int i = lane; i < NBINS; i += 32) hf[i] = (float)ghist[i];
    __syncthreads();

    const int m  = lane & 15;                // row (A) / column (B,D) owned by lane
    const int hi = lane >> 4;                // 0: lanes 0-15, 1: lanes 16-31

    v8f acc = {};
    #pragma unroll
    for (int j = 0; j < 4; ++j) {
        // A 16x4 f32 layout: VGPR0 = K 4j (lanes0-15) / 4j+2 (lanes16-31), VGPR1 = +1
        const int k0 = 4 * j + 2 * hi;
        const int k1 = k0 + 1;
        v2f a, b;
        a.x = (k0 <= m) ? 1.0f : 0.0f;       // L[m][k0]
        a.y = (k1 <= m) ? 1.0f : 0.0f;       // L[m][k1]
        // B 4x16 f32: rows striped across lanes, same K split as A
        b.x = hf[m * 16 + k0];               // H[k0][col=m]
        b.y = hf[m * 16 + k1];               // H[k1][col=m]
        acc = __builtin_amdgcn_wmma_f32_16x16x4_f32(
            /*neg_a=*/false, a, /*neg_b=*/false, b,
            /*c_mod=*/(short)0, acc, /*reuse_a=*/false, /*reuse_b=*/false);
    }
    // D layout: lane holds column m, rows v + 8*hi -> bin = m*16 + r
    #pragma unroll
    for (int v = 0; v < 8; ++v) S[m * 16 + (v + 8 * hi)] = acc[v];
    __syncthreads();

    // Each lane owns 8 consecutive bins: all in segment seg = lane>>1.
    const int seg = lane >> 1;
    float off = 0.0f, grand = 0.0f;
    for (int s2 = 0; s2 < 16; ++s2) {
        float tot = S[s2 * 16 + 15];
        if (s2 < seg) off += tot;
        grand += tot;
    }

    // Target normal CDF over linspace(-1,1,256), mean 0 std 0.2
    for (int i = lane; i < NBINS; i += 32) {
        float x = -1.0f + (2.0f / 255.0f) * (float)i;
        float z = x * 5.0f;                                   // /0.2
        nc[i] = 0.5f * (1.0f + erff(z * 0.70710678118654752f));
    }
    __syncthreads();
    float nc255 = nc[255];
    __syncthreads();
    for (int i = lane; i < NBINS; i += 32) nc[i] = nc[i] / nc255;
    __syncthreads();

    // searchsorted(nc, src_cdf[b], 'left') == count(nc[i] < v); fold rescale in.
    #pragma unroll
    for (int k = 0; k < 8; ++k) {
        int b = lane * 8 + k;
        float sc = (S[b] + off) / grand;                      // normalized src CDF
        int idx = 0;
        for (int i = 0; i < NBINS; ++i) idx += (nc[i] < sc) ? 1 : 0;
        lut[b] = (float)idx / 255.0f * 2.0f - 1.0f;
    }
}

// ---------------------------------------------------------------------------
// Kernel 3: apply. Stage the 1KB LUT into LDS with the CDNA5 async path
// (global_load_async_to_lds_b128 + s_wait_asynccnt), then stream float4 in,
// 4 LDS gathers, float4 out.
// ---------------------------------------------------------------------------
__global__ void __launch_bounds__(256) apply_kernel(const float* __restrict__ src,
                                                    const float* __restrict__ lut,
                                                    float* __restrict__ out,
                                                    int n4) {
    __shared__ float lut_s[NBINS];
    const int t = threadIdx.x;

    if (t < 64) {  // 64 lanes x 16B = 1024B = full LUT
        uint32_t lds_off = (uint32_t)(uintptr_t)(&lut_s[t * 4]);   // low 32b of flat = LDS offset
        uint64_t gaddr   = (uint64_t)(uintptr_t)(lut + t * 4);
        asm volatile("global_load_async_to_lds_b128 %0, %1, off"
                     :: "v"(lds_off), "v"(gaddr) : "memory");
    }
    asm volatile("s_wait_asynccnt 0x0" ::: "memory");
    __syncthreads();

    const float4* s4 = (const float4*)src;
    float4*       o4 = (float4*)out;
    const int stride = gridDim.x * 256;

    for (int i = blockIdx.x * 256 + t; i < n4; i += stride) {
        int ip = i + stride;
        if (ip < n4) __builtin_prefetch(&s4[ip], 0, 0);
        float4 v = s4[i];
        float4 r;
        r.x = lut_s[bin_of(v.x)];
        r.y = lut_s[bin_of(v.y)];
        r.z = lut_s[bin_of(v.z)];
        r.w = lut_s[bin_of(v.w)];
        o4[i] = r;
    }
}

// ---------------------------------------------------------------------------
extern "C" void kernel_launch(void* const* d_in, const int* in_sizes, int n_in,
                              void* d_out, int out_size, void* d_ws, size_t ws_size,
                              hipStream_t stream) {
    const float* src = (const float*)d_in[0];
    float*       out = (float*)d_out;

    unsigned int* ghist = (unsigned int*)d_ws;                       // 256 * u32
    float*        lut   = (float*)((char*)d_ws + NBINS * sizeof(unsigned int)); // 256 * f32

    const int n  = in_sizes[0];      // 8192*8192 = 67,108,864
    const int n4 = n >> 2;           // float4 count

    hist_zero_kernel<<<1, 256, 0, stream>>>(ghist);
    hist_kernel<<<2048, 256, 0, stream>>>(src, ghist, n4);
    scan_lut_kernel<<<1, 32, 0, stream>>>(ghist, lut);
    apply_kernel<<<2048, 256, 0, stream>>>(src, lut, out, n4);
}